// ContentEncoder_2121713844277
// MI455X (gfx1250) — compile-verified
//
#include <hip/hip_runtime.h>
#include <hip/hip_bf16.h>

typedef __attribute__((ext_vector_type(2))) float v2f;
typedef __attribute__((ext_vector_type(8))) float v8f;

// ---------------------------------------------------------------------------
// GEMM: out[b][m][p] = sum_k A[m][k] * Col[k][b*P+p] + bias[m]
// fp32 WMMA 16x16x4. Each wave computes a 32x64 register tile (2x4 subtiles,
// 8 v_wmma per k-step). 4 waves/block -> block tile 32 rows x 256 cols.
//  - Ap: weights packed+zero-padded to [Mp][Kd4] (Mp mult of 32, Kd4 mult of 4)
//  - Bq: k-quad packed columns: element (k,n) at ((k>>2)*N + n)*4 + (k&3),
//        rows [Kd,Kd4) zeroed. All fragment loads are aligned v2f (b64).
//  - N % 256 == 0, P = 1<<pshift (output scattered into NCHW).
// ---------------------------------------------------------------------------
__global__ __launch_bounds__(128) void gemm_bias_wmma(
    const float* __restrict__ Ap, const float* __restrict__ Bq,
    const float* __restrict__ bias, float* __restrict__ Cm,
    int M, int N, int Kd4, int pshift)
{
  const int lane = threadIdx.x & 31;
  const int wave = threadIdx.x >> 5;
  const int half = lane >> 4;
  const int l    = lane & 15;
  const int colw = blockIdx.x * 256 + wave * 64;  // wave's first column
  const int row0 = blockIdx.y * 32;               // wave's first row

  const float* arow0 = Ap + (size_t)(row0 + l) * Kd4 + 2 * half;
  const float* arow1 = arow0 + (size_t)16 * Kd4;
  const float* bbase = Bq + (size_t)(colw + l) * 4 + 2 * half;
  const size_t bstep = (size_t)N * 4;             // one k-quad of B

  v8f acc[2][4] = {};

  for (int kb = 0; kb < Kd4; kb += 4) {
    const v2f a0 = *(const v2f*)(arow0 + kb);
    const v2f a1 = *(const v2f*)(arow1 + kb);
    const float* bq = bbase + (size_t)(kb >> 2) * bstep;
    v2f bf[4];
#pragma unroll
    for (int nt = 0; nt < 4; ++nt)
      bf[nt] = *(const v2f*)(bq + nt * 64);
#pragma unroll
    for (int nt = 0; nt < 4; ++nt)
      acc[0][nt] = __builtin_amdgcn_wmma_f32_16x16x4_f32(
          false, a0, false, bf[nt], (short)0, acc[0][nt], false, false);
#pragma unroll
    for (int nt = 0; nt < 4; ++nt)
      acc[1][nt] = __builtin_amdgcn_wmma_f32_16x16x4_f32(
          false, a1, false, bf[nt], (short)0, acc[1][nt], false, false);
  }

  // Scatter into NCHW: column n = b*P + p with P = 1<<pshift.
  const int pmask = (1 << pshift) - 1;
#pragma unroll
  for (int mt = 0; mt < 2; ++mt) {
    const int rb = row0 + mt * 16;        // subtile rows [rb, rb+16)
    if (rb + 16 <= M) {                   // uniform fast path (M=64/128/256)
#pragma unroll
      for (int nt = 0; nt < 4; ++nt) {
        const int bc   = colw + nt * 16 + l;
        const int bimg = bc >> pshift;
        const int p    = bc & pmask;
        float* dst = Cm + (((size_t)(bimg * M + rb + 8 * half)) << pshift) + p;
#pragma unroll
        for (int r = 0; r < 8; ++r)
          dst[(size_t)r << pshift] = acc[mt][nt][r] + bias[rb + 8 * half + r];
      }
    } else if (rb < M) {                  // straddling tile (M=147 only)
#pragma unroll
      for (int nt = 0; nt < 4; ++nt) {
        const int bc   = colw + nt * 16 + l;
        const int bimg = bc >> pshift;
        const int p    = bc & pmask;
#pragma unroll
        for (int r = 0; r < 8; ++r) {
          const int crow = rb + r + 8 * half;
          if (crow < M)
            Cm[(((size_t)(bimg * M + crow)) << pshift) + p] = acc[mt][nt][r] + bias[crow];
        }
      }
    }
  }
}

// ---------------------------------------------------------------------------
// Pack weights [M][Kd] -> zero-padded [Mp][Kd4].
// ---------------------------------------------------------------------------
__global__ __launch_bounds__(256) void pack_A_k(
    const float* __restrict__ A, float* __restrict__ Ap,
    int M, int Kd, int Mp, int Kd4)
{
  const long total = (long)Mp * Kd4;
  const long i = (long)blockIdx.x * blockDim.x + threadIdx.x;
  if (i >= total) return;
  const int r = (int)(i / Kd4);
  const int k = (int)(i % Kd4);
  Ap[i] = (r < M && k < Kd) ? A[(size_t)r * Kd + k] : 0.0f;
}

// ---------------------------------------------------------------------------
// Zero the K-tail rows [Kd, Kd4) of a k-quad packed column matrix.
// ---------------------------------------------------------------------------
__global__ __launch_bounds__(256) void pad_col_k(
    float* __restrict__ col, int Kd, int Kd4, int NP)
{
  const long total = (long)(Kd4 - Kd) * NP;
  const long i = (long)blockIdx.x * blockDim.x + threadIdx.x;
  if (i >= total) return;
  const int rk = Kd + (int)(i / NP);
  const int n  = (int)(i % NP);
  col[((size_t)(rk >> 2) * NP + n) * 4 + (rk & 3)] = 0.0f;
}

// ---------------------------------------------------------------------------
// im2col for a regular conv (zero pad or reflect pad, NCHW input), writing
// the k-quad packed column layout.
// ---------------------------------------------------------------------------
__global__ __launch_bounds__(256) void im2col_k(
    const float* __restrict__ x, float* __restrict__ col,
    int Bn, int Cc, int H, int W, int kh, int kw, int stride, int pad,
    int Ho, int Wo, int reflect)
{
  const long total = (long)Cc * kh * kw * Bn * Ho * Wo;
  const long i = (long)blockIdx.x * blockDim.x + threadIdx.x;
  if (i >= total) return;
  const int P  = Ho * Wo;
  const int NP = Bn * P;
  const int n  = (int)(i % NP);
  const int rk = (int)(i / NP);
  const int K  = kh * kw;
  const int c  = rk / K;
  const int k  = rk % K;
  const int ky = k / kw, kx = k % kw;
  const int b  = n / P;
  const int p  = n % P;
  const int ho = p / Wo, wo = p % Wo;
  int iy = ho * stride - pad + ky;
  int ix = wo * stride - pad + kx;
  float v = 0.0f;
  if (reflect) {
    iy = iy < 0 ? -iy : (iy >= H ? 2 * H - 2 - iy : iy);
    ix = ix < 0 ? -ix : (ix >= W ? 2 * W - 2 - ix : ix);
    v = x[((size_t)(b * Cc + c) * H + iy) * W + ix];
  } else if (iy >= 0 && iy < H && ix >= 0 && ix < W) {
    v = x[((size_t)(b * Cc + c) * H + iy) * W + ix];
  }
  col[((size_t)(rk >> 2) * NP + n) * 4 + (rk & 3)] = v;
}

// ---------------------------------------------------------------------------
// DCNv2 column build: bilinear sample at (base + offset), times sigmoid(mask).
// om is NCHW [B][3K][Ho][Wo]; off_y = ch 2k, off_x = ch 2k+1, mask = ch 2K+k.
// One thread per (b,k,p); loops over input channels. k-quad packed output.
// ---------------------------------------------------------------------------
__global__ __launch_bounds__(256) void dcn_cols_k(
    const float* __restrict__ x, const float* __restrict__ om,
    float* __restrict__ col,
    int Bn, int Cc, int H, int W, int kh, int kw, int stride, int pad,
    int Ho, int Wo)
{
  const int K = kh * kw;
  const int P = Ho * Wo;
  const long total = (long)Bn * K * P;
  const long i = (long)blockIdx.x * blockDim.x + threadIdx.x;
  if (i >= total) return;
  const int p = (int)(i % P);
  const int t = (int)(i / P);
  const int k = t % K;
  const int b = t / K;
  const int ho = p / Wo, wo = p % Wo;
  const int ky = k / kw, kx = k % kw;

  const size_t omb = (size_t)b * (3 * K) * P;
  const float offy = om[omb + (size_t)(2 * k) * P + p];
  const float offx = om[omb + (size_t)(2 * k + 1) * P + p];
  float m = om[omb + (size_t)(2 * K + k) * P + p];
  m = 1.0f / (1.0f + expf(-m));

  const float py = (float)(ho * stride - pad + ky) + offy;
  const float px = (float)(wo * stride - pad + kx) + offx;
  const float y0f = floorf(py), x0f = floorf(px);
  const float dy = py - y0f, dx = px - x0f;
  const int y0 = (int)y0f, x0 = (int)x0f;
  const int y1 = y0 + 1,   x1 = x0 + 1;
  const bool vy0 = (y0 >= 0 && y0 < H), vy1 = (y1 >= 0 && y1 < H);
  const bool vx0 = (x0 >= 0 && x0 < W), vx1 = (x1 >= 0 && x1 < W);
  const int cy0 = min(max(y0, 0), H - 1), cy1 = min(max(y1, 0), H - 1);
  const int cx0 = min(max(x0, 0), W - 1), cx1 = min(max(x1, 0), W - 1);
  const float w00 = (1.0f - dy) * (1.0f - dx) * ((vy0 && vx0) ? m : 0.0f);
  const float w01 = (1.0f - dy) * dx          * ((vy0 && vx1) ? m : 0.0f);
  const float w10 = dy * (1.0f - dx)          * ((vy1 && vx0) ? m : 0.0f);
  const float w11 = dy * dx                   * ((vy1 && vx1) ? m : 0.0f);

  const size_t NP = (size_t)Bn * P;
  const size_t cn = (size_t)b * P + p;
  const int i00 = cy0 * W + cx0, i01 = cy0 * W + cx1;
  const int i10 = cy1 * W + cx0, i11 = cy1 * W + cx1;
  for (int c = 0; c < Cc; ++c) {
    const float* xc = x + (size_t)(b * Cc + c) * H * W;
    const float v = w00 * xc[i00] + w01 * xc[i01] + w10 * xc[i10] + w11 * xc[i11];
    const int rk = c * K + k;
    col[((size_t)(rk >> 2) * NP + cn) * 4 + (rk & 3)] = v;
  }
}

// ---------------------------------------------------------------------------
// Instance norm over H*W per (b,c) channel; optional relu; optional residual.
// grid.x = B*C; dst = (res? res + : ) inorm(y) [relu]
// ---------------------------------------------------------------------------
__global__ __launch_bounds__(256) void inorm_k(
    const float* __restrict__ y, const float* __restrict__ res,
    float* __restrict__ dst, int HW, int relu)
{
  const size_t base = (size_t)blockIdx.x * HW;
  __shared__ float ssum[256];
  __shared__ float ssq[256];
  float s = 0.0f, q = 0.0f;
  for (int i = threadIdx.x; i < HW; i += 256) {
    const float v = y[base + i];
    s += v; q += v * v;
  }
  ssum[threadIdx.x] = s; ssq[threadIdx.x] = q;
  __syncthreads();
  for (int off = 128; off > 0; off >>= 1) {
    if (threadIdx.x < off) {
      ssum[threadIdx.x] += ssum[threadIdx.x + off];
      ssq[threadIdx.x]  += ssq[threadIdx.x + off];
    }
    __syncthreads();
  }
  const float mean = ssum[0] / (float)HW;
  const float var  = ssq[0] / (float)HW - mean * mean;
  const float inv  = rsqrtf(var + 1e-5f);
  for (int i = threadIdx.x; i < HW; i += 256) {
    float v = (y[base + i] - mean) * inv;
    if (relu) v = fmaxf(v, 0.0f);
    if (res)  v += res[base + i];
    dst[base + i] = v;
  }
}

// ---------------------------------------------------------------------------
// Host orchestration
// ---------------------------------------------------------------------------
static inline void run_gemm(const float* A, float* col, float* apk,
                            const float* bias, float* C,
                            int M, int N, int Kd, int P, hipStream_t s)
{
  const int Kd4 = (Kd + 3) & ~3;
  const int Mp  = (M + 31) & ~31;
  {
    const long n = (long)Mp * Kd4;
    pack_A_k<<<(int)((n + 255) / 256), 256, 0, s>>>(A, apk, M, Kd, Mp, Kd4);
  }
  if (Kd4 != Kd) {
    const long n = (long)(Kd4 - Kd) * N;
    pad_col_k<<<(int)((n + 255) / 256), 256, 0, s>>>(col, Kd, Kd4, N);
  }
  int pshift = 0;
  while ((1 << pshift) < P) ++pshift;
  dim3 grid(N / 256, Mp / 32);
  gemm_bias_wmma<<<grid, 128, 0, s>>>(apk, col, bias, C, M, N, Kd4, pshift);
}

static inline void run_im2col(const float* x, float* col, int B, int C, int H, int W,
                              int kh, int kw, int stride, int pad, int Ho, int Wo,
                              int reflect, hipStream_t s)
{
  const long total = (long)C * kh * kw * B * Ho * Wo;
  const int blocks = (int)((total + 255) / 256);
  im2col_k<<<blocks, 256, 0, s>>>(x, col, B, C, H, W, kh, kw, stride, pad, Ho, Wo, reflect);
}

static inline void run_dcn(const float* x, const float* om, float* col,
                           int B, int C, int H, int W, int kh, int kw,
                           int stride, int pad, int Ho, int Wo, hipStream_t s)
{
  const long total = (long)B * kh * kw * Ho * Wo;
  const int blocks = (int)((total + 255) / 256);
  dcn_cols_k<<<blocks, 256, 0, s>>>(x, om, col, B, C, H, W, kh, kw, stride, pad, Ho, Wo);
}

extern "C" void kernel_launch(void* const* d_in, const int* in_sizes, int n_in,
                              void* d_out, int out_size, void* d_ws, size_t ws_size,
                              hipStream_t stream)
{
  (void)in_sizes; (void)n_in; (void)out_size; (void)ws_size;
  const float* x    = (const float*)d_in[0];
  const float* w1   = (const float*)d_in[1];
  const float* b1   = (const float*)d_in[2];
  const float* wo1  = (const float*)d_in[3];
  const float* bo1  = (const float*)d_in[4];
  const float* w2   = (const float*)d_in[5];
  const float* b2   = (const float*)d_in[6];
  const float* wo2  = (const float*)d_in[7];
  const float* bo2  = (const float*)d_in[8];
  const float* w3   = (const float*)d_in[9];
  const float* b3   = (const float*)d_in[10];
  const float* wo3  = (const float*)d_in[11];
  const float* bo3  = (const float*)d_in[12];
  const float* r1aw = (const float*)d_in[13];
  const float* r1ab = (const float*)d_in[14];
  const float* r1bw = (const float*)d_in[15];
  const float* r1bb = (const float*)d_in[16];
  const float* r2aw = (const float*)d_in[17];
  const float* r2ab = (const float*)d_in[18];
  const float* r2bw = (const float*)d_in[19];
  const float* r2bb = (const float*)d_in[20];

  // Output tuple (h, skip1, skip2) flattened in order.
  float* out_h  = (float*)d_out;                      // [4,256,32,32]
  float* out_s1 = out_h + (size_t)4 * 256 * 32 * 32;  // [4,64,128,128]
  float* out_s2 = out_s1 + (size_t)4 * 64 * 128 * 128;// [4,128,64,64]

  // Workspace layout (floats): total ~34.8M fl = ~139 MB
  float* ws  = (float*)d_ws;
  float* COL = ws;                 // 16,777,216  (max: stage-2 cols 1024x16384)
  float* OM  = COL + 16777216;     //  9,633,792  (max: stage-1 offsets 4x147x128x128)
  float* Y   = OM  + 9633792;      //  4,194,304  (max: stage-1 pre-norm 4x64x128x128)
  float* H3  = Y   + 4194304;      //  1,048,576  (stage-3 activation)
  float* T   = H3  + 1048576;      //  1,048,576  (resblock temp)
  float* R1  = T   + 1048576;      //  1,048,576  (resblock-1 output)
  float* APK = R1  + 1048576;      //  1,048,576  (packed weights, max 256x2304)

  const int B = 4;

  // ---- Stage 1: mdcn 3->64, 7x7, s1 p3, 128x128 -> skip1 ----
  run_im2col(x, COL, B, 3, 128, 128, 7, 7, 1, 3, 128, 128, 0, stream);
  run_gemm(wo1, COL, APK, bo1, OM, 147, 65536, 147, 16384, stream);   // offset conv
  run_dcn(x, OM, COL, B, 3, 128, 128, 7, 7, 1, 3, 128, 128, stream);
  run_gemm(w1, COL, APK, b1, Y, 64, 65536, 147, 16384, stream);       // deformable conv
  inorm_k<<<B * 64, 256, 0, stream>>>(Y, nullptr, out_s1, 16384, 1);

  // ---- Stage 2: mdcn 64->128, 4x4, s2 p1, ->64x64 -> skip2 ----
  run_im2col(out_s1, COL, B, 64, 128, 128, 4, 4, 2, 1, 64, 64, 0, stream);
  run_gemm(wo2, COL, APK, bo2, OM, 48, 16384, 1024, 4096, stream);
  run_dcn(out_s1, OM, COL, B, 64, 128, 128, 4, 4, 2, 1, 64, 64, stream);
  run_gemm(w2, COL, APK, b2, Y, 128, 16384, 1024, 4096, stream);
  inorm_k<<<B * 128, 256, 0, stream>>>(Y, nullptr, out_s2, 4096, 1);

  // ---- Stage 3: mdcn 128->256, 4x4, s2 p1, ->32x32 -> H3 ----
  run_im2col(out_s2, COL, B, 128, 64, 64, 4, 4, 2, 1, 32, 32, 0, stream);
  run_gemm(wo3, COL, APK, bo3, OM, 48, 4096, 2048, 1024, stream);
  run_dcn(out_s2, OM, COL, B, 128, 64, 64, 4, 4, 2, 1, 32, 32, stream);
  run_gemm(w3, COL, APK, b3, Y, 256, 4096, 2048, 1024, stream);
  inorm_k<<<B * 256, 256, 0, stream>>>(Y, nullptr, H3, 1024, 1);

  // ---- Resblock 1: R1 = H3 + inorm(conv(relu(inorm(conv(rpad(H3)))))) ----
  run_im2col(H3, COL, B, 256, 32, 32, 3, 3, 1, 1, 32, 32, 1, stream);
  run_gemm(r1aw, COL, APK, r1ab, Y, 256, 4096, 2304, 1024, stream);
  inorm_k<<<B * 256, 256, 0, stream>>>(Y, nullptr, T, 1024, 1);
  run_im2col(T, COL, B, 256, 32, 32, 3, 3, 1, 1, 32, 32, 1, stream);
  run_gemm(r1bw, COL, APK, r1bb, Y, 256, 4096, 2304, 1024, stream);
  inorm_k<<<B * 256, 256, 0, stream>>>(Y, H3, R1, 1024, 0);

  // ---- Resblock 2: -> out_h ----
  run_im2col(R1, COL, B, 256, 32, 32, 3, 3, 1, 1, 32, 32, 1, stream);
  run_gemm(r2aw, COL, APK, r2ab, Y, 256, 4096, 2304, 1024, stream);
  inorm_k<<<B * 256, 256, 0, stream>>>(Y, nullptr, T, 1024, 1);
  run_im2col(T, COL, B, 256, 32, 32, 3, 3, 1, 1, 32, 32, 1, stream);
  run_gemm(r2bw, COL, APK, r2bb, Y, 256, 4096, 2304, 1024, stream);
  inorm_k<<<B * 256, 256, 0, stream>>>(Y, R1, out_h, 1024, 0);
}